// ImageTransformTorch_25744033972487
// MI455X (gfx1250) — compile-verified
//
#include <hip/hip_runtime.h>

// Problem constants (from the reference): xs = (B=4, H=224, W=224, C=3) fp32.
// Stage 1: S=21 half-pixel shifts  -> mid (S*B, H, W, C)   ~50.6 MB (d_ws, L2-resident)
// Stage 2: R=5 rotations of mid    -> out (R*S*B, H, W, C) ~253 MB (NT stores)
#define HW     224
#define CH     3
#define BATCH  4
#define NSHIFT 21
#define NROT   5
#define ROWS_PER_BLOCK 4
#define ROW_ELEMS (HW * CH)            // 672 floats per image row

// Shift table in HALF-PIXEL units. Reference order: up p=1..5, down p=1..5,
// left p=1..5, right p=1..5 then p=0.  ix = x + dx2*0.5, iy = y + dy2*0.5.
__constant__ int SHIFT_DX2[NSHIFT] = { 0, 0, 0, 0, 0,   0, 0, 0, 0, 0,
                                       1, 2, 3, 4, 5,  -1,-2,-3,-4,-5,  0 };
__constant__ int SHIFT_DY2[NSHIFT] = { 1, 2, 3, 4, 5,  -1,-2,-3,-4,-5,
                                       0, 0, 0, 0, 0,   0, 0, 0, 0, 0,  0 };

// cos/sin of ANGLES = [-0.5236, -0.2618, 0.0, 0.2618, 0.5236] (double-evaluated, fp32-rounded)
__constant__ float ROT_C[NROT] = { 0.86602479f, 0.96592567f, 1.0f, 0.96592567f, 0.86602479f };
__constant__ float ROT_S[NROT] = { -0.50000106f, -0.25881964f, 0.0f, 0.25881964f, 0.50000106f };

typedef unsigned int v4u __attribute__((ext_vector_type(4)));
typedef int          v8i __attribute__((ext_vector_type(8)));
typedef int          v4i __attribute__((ext_vector_type(4)));

// ------------------------------------------------------------------------------------
// Stage 1: half-pixel shifts. One block per (s, b, 4-row output band).
// The <=5 needed source rows are a contiguous 2D band -> DMA'd into LDS with the
// Tensor Data Mover (D# per CDNA5 ISA), then the
// axis-aligned (<=2 taps per axis) interpolation reads LDS.
// ------------------------------------------------------------------------------------
__global__ __launch_bounds__(HW)
void stage1_shift(const float* __restrict__ xs, float* __restrict__ mid)
{
    __shared__ float tile[(ROWS_PER_BLOCK + 1) * ROW_ELEMS];   // 5 rows * 672 f32 = 13.4 KB, LDS offset 0

    const int x      = threadIdx.x;                 // 224 threads = 7 waves
    int t            = blockIdx.x;
    const int yband  = t % (HW / ROWS_PER_BLOCK);   // 56 bands
    t               /= (HW / ROWS_PER_BLOCK);
    const int b      = t % BATCH;
    const int s      = t / BATCH;
    const int y_base = yband * ROWS_PER_BLOCK;

    const int dx2 = SHIFT_DX2[s];
    const int dy2 = SHIFT_DY2[s];

    // Uniform source-row band for the 4 output rows: iy = y + dy2/2.
    const int r_lo = (int)floorf((float)y_base + 0.5f * (float)dy2);
    const int r_hi = (int)floorf((float)(y_base + ROWS_PER_BLOCK - 1) + 0.5f * (float)dy2)
                     + ((dy2 & 1) ? 1 : 0);
    const int cr0   = max(r_lo, 0);
    const int cr1   = min(r_hi, HW - 1);
    const int nrows = cr1 - cr0 + 1;                // 2 to 5 here (never empty for |shift|<=2.5px)

    if (nrows > 0) {
#if __has_builtin(__builtin_amdgcn_tensor_load_to_lds)
        if (threadIdx.x < 32) {                     // wave 0 issues the DMA (TDM ignores EXEC)
            const unsigned long long ga =
                (unsigned long long)(const void*)(xs + ((size_t)b * HW + cr0) * ROW_ELEMS);
            // ---- D# group 0: count=1 | lds_addr=0 | global_addr | type=2 ("image") ----
            v4u g0;
            g0[0] = 1u;                                             // count=1, user descriptor
            g0[1] = 0u;                                             // lds_addr: tile[] is at LDS offset 0
            g0[2] = (unsigned)(ga & 0xffffffffull);                 // global_addr[31:0]
            g0[3] = (unsigned)((ga >> 32) & 0x01ffffffull)          // global_addr[56:32]
                    | (2u << 30);                                   // type=2
            // ---- D# group 1: 2D tile, data_size=4B, tile = ROW_ELEMS x nrows, stride = ROW_ELEMS ----
            v8i g1;
            g1[0] = (int)(2u << 16);                                // wg_mask=0, data_size=4B, no flags
            g1[1] = (int)(((unsigned)ROW_ELEMS & 0xffffu) << 16);   // tensor_dim0[15:0]
            g1[2] = (int)((ROW_ELEMS >> 16) | (((unsigned)nrows & 0xffffu) << 16)); // td0 hi | td1 lo
            g1[3] = (int)(((unsigned)ROW_ELEMS & 0xffffu) << 16);   // td1 hi (0) | tile_dim0 = 672
            g1[4] = (int)((unsigned)nrows & 0xffffu);               // tile_dim1 = nrows, tile_dim2 = 0
            g1[5] = ROW_ELEMS;                                      // tensor_dim0_stride low32
            g1[6] = 0;                                              // stride hi16 | td1_stride lo16
            g1[7] = 0;                                              // td1_stride hi32 (2D: unused)
            v4i g2 = {0, 0, 0, 0};                                  // 2D tensor: groups 2/3 zero
            v4i g3 = {0, 0, 0, 0};
            v8i g4 = {0, 0, 0, 0, 0, 0, 0, 0};                      // extra arg of 6-arg form (unused, zeros)
            __builtin_amdgcn_tensor_load_to_lds(g0, g1, g2, g3, g4, 0);
#if __has_builtin(__builtin_amdgcn_s_wait_tensorcnt)
            __builtin_amdgcn_s_wait_tensorcnt(0);
#else
            asm volatile("s_wait_tensorcnt 0x0" ::: "memory");
#endif
        }
#else
        // Fallback: cooperative vector-path copy into LDS.
        const float* __restrict__ src = xs + ((size_t)b * HW + cr0) * ROW_ELEMS;
        for (int i = threadIdx.x; i < nrows * ROW_ELEMS; i += HW) tile[i] = src[i];
#endif
    }
    // The TDM builtin carries no pointer operand, so clang cannot see that it writes
    // tile[] and would otherwise fold the (apparently never-stored) LDS loads away.
    // Escape the array into an opaque asm with a memory clobber: zero instructions,
    // but the compiler must now assume tile[] was written.
    asm volatile("" :: "v"(&tile[0]) : "memory");
    __syncthreads();

    // Horizontal taps are uniform per block: ix = x + dx2/2.
    const float ix = (float)x + 0.5f * (float)dx2;
    const int   x0 = (int)floorf(ix);
    const float wx = ix - (float)x0;

    #pragma unroll
    for (int j = 0; j < ROWS_PER_BLOCK; ++j) {
        const int   y  = y_base + j;
        const float iy = (float)y + 0.5f * (float)dy2;
        const int   y0 = (int)floorf(iy);
        const float wy = iy - (float)y0;

        float a0 = 0.f, a1 = 0.f, a2 = 0.f;
        #pragma unroll
        for (int dy = 0; dy < 2; ++dy) {
            const int   rr  = y0 + dy;
            const float wyv = dy ? wy : (1.f - wy);
            if (wyv == 0.f || rr < 0 || rr >= HW) continue;      // zero pad / zero weight
            const float* __restrict__ lrow = tile + (size_t)(rr - cr0) * ROW_ELEMS;
            #pragma unroll
            for (int dx = 0; dx < 2; ++dx) {
                const int   xx = x0 + dx;
                const float w  = (dx ? wx : (1.f - wx)) * wyv;
                if (w == 0.f || xx < 0 || xx >= HW) continue;
                const float* __restrict__ p = lrow + xx * CH;
                a0 += w * p[0];  a1 += w * p[1];  a2 += w * p[2];
            }
        }

        float* __restrict__ o = mid + (((size_t)(s * BATCH + b) * HW + y) * HW + x) * CH;
        o[0] = a0;  o[1] = a1;  o[2] = a2;                       // RT stores: keep mid in L2
    }
}

// ------------------------------------------------------------------------------------
// Stage 2: rotation (full 4-tap bilinear) — bandwidth-dominant pass.
// Reads the L2-resident mid tensor; writes 253 MB with NT stores so the output
// stream does not evict mid from the 192 MB L2.
// ------------------------------------------------------------------------------------
__global__ __launch_bounds__(256)
void stage2_rotate(const float* __restrict__ mid, float* __restrict__ out, int total)
{
    int idx = blockIdx.x * blockDim.x + threadIdx.x;
    if (idx >= total) return;                       // total = NROT*NSHIFT*BATCH*HW*HW pixels

    int x  = idx % HW;
    int t  = idx / HW;
    int y  = t % HW;  t /= HW;
    int n1 = t % (NSHIFT * BATCH);                  // shifted-image index s*B+b
    int r  = t / (NSHIFT * BATCH);

    float c = ROT_C[r];
    float s = ROT_S[r];

    // align_corners=True affine rotation in pixel space (H == W so cx == cy)
    const float ctr = (float)(HW - 1) * 0.5f;
    float u  = (float)x - ctr;
    float v  = (float)y - ctr;
    float ix =  c * u + s * v + ctr;
    float iy = -s * u + c * v + ctr;

    int   x0 = (int)floorf(ix);  float wx = ix - (float)x0;
    int   y0 = (int)floorf(iy);  float wy = iy - (float)y0;

    float a0 = 0.f, a1 = 0.f, a2 = 0.f;
    const float* __restrict__ img = mid + (size_t)n1 * (HW * HW * CH);

    #pragma unroll
    for (int dy = 0; dy < 2; ++dy) {
        int   yy  = y0 + dy;
        float wyv = dy ? wy : (1.f - wy);
        if (yy < 0 || yy >= HW) continue;           // zero padding
        #pragma unroll
        for (int dx = 0; dx < 2; ++dx) {
            int   xx = x0 + dx;
            float w  = (dx ? wx : (1.f - wx)) * wyv;
            if (w == 0.f || xx < 0 || xx >= HW) continue;
            const float* __restrict__ p = img + ((size_t)yy * HW + xx) * CH;
            a0 += w * p[0];  a1 += w * p[1];  a2 += w * p[2];
        }
    }

    float* __restrict__ o = out + (size_t)idx * CH;
    __builtin_nontemporal_store(a0, o + 0);
    __builtin_nontemporal_store(a1, o + 1);
    __builtin_nontemporal_store(a2, o + 2);
}

extern "C" void kernel_launch(void* const* d_in, const int* in_sizes, int n_in,
                              void* d_out, int out_size, void* d_ws, size_t ws_size,
                              hipStream_t stream)
{
    (void)in_sizes; (void)n_in; (void)out_size; (void)ws_size;

    const float* xs  = (const float*)d_in[0];   // (4, 224, 224, 3) fp32, NHWC
    float*       out = (float*)d_out;           // (105, 4, 224, 224, 3) fp32
    float*       mid = (float*)d_ws;            // (84, 224, 224, 3) fp32 = 50.6 MB scratch

    const int blocks1 = NSHIFT * BATCH * (HW / ROWS_PER_BLOCK);   // 4704 row-band blocks
    const int total2  = NROT * NSHIFT * BATCH * HW * HW;          // 21,073,920 pixels

    stage1_shift<<<blocks1, HW, 0, stream>>>(xs, mid);
    stage2_rotate<<<(total2 + 255) / 256, 256, 0, stream>>>(mid, out, total2);
}